// Pool_layer_37873021616801
// MI455X (gfx1250) — compile-verified
//
#include <hip/hip_runtime.h>
#include <hip/hip_bf16.h>

typedef __attribute__((ext_vector_type(2))) float v2f;
typedef __attribute__((ext_vector_type(8))) float v8f;

#define BB   4
#define NN   8192
#define CC   256
#define QQ   2048           // NN / POOLING_RATE
#define QT   16             // queries per wave (one WMMA tile of columns)

// ---------------------------------------------------------------------------
// Kernel 1: ball-query via V_WMMA_F32_16X16X4_F32, 2 tiles (32 candidates)
// per iteration. One wave handles 16 sampled query points of one batch.
//  A (M=16 candidates x K=4) : [cx, cy, cz, |c|^2]
//  B (K=4 x N=16 queries)    : [-2qx, -2qy, -2qz, 1]
//  C (preloaded)             : |q|^2 per query column
//  => D[m][n] = |c_m - q_n|^2 in a single WMMA, no post-processing.
// Lane n / n+16 own query column n; per 32 candidates we build one 32-bit
// in-index-order "in ball" mask and take its first 4 set bits (== jnp.sort of
// index-or-N, take nsample). Wave-uniform early exit when all 16 queries done.
// ---------------------------------------------------------------------------
__global__ __launch_bounds__(32) void qball_wmma_kernel(
    const float* __restrict__ vertices,   // BB*NN*3
    const int*   __restrict__ radius_p,   // scalar (int)
    const int*   __restrict__ sample_idx, // QQ
    int*         __restrict__ nbr,        // BB*QQ*4  (workspace)
    float*       __restrict__ out_verts)  // BB*QQ*3  (vertices_pool)
{
    const int lane  = threadIdx.x;        // 0..31
    const int g     = blockIdx.x;         // 0 .. BB*QQ/QT - 1
    const int b     = g / (QQ / QT);
    const int jbase = (g % (QQ / QT)) * QT;
    const int q     = lane & 15;
    const bool hi   = lane >= 16;

    const float r  = (float)radius_p[0];
    const float r2 = r * r;

    const float* vb = vertices + (size_t)b * NN * 3;

    // Gather this lane's query point (column q of the B operand).
    const int   j   = jbase + q;
    const int   qg  = sample_idx[j];
    const float qx  = vb[qg * 3 + 0];
    const float qy  = vb[qg * 3 + 1];
    const float qz  = vb[qg * 3 + 2];
    const float qsq = qx * qx + qy * qy + qz * qz;

    // B operand: VGPR0 = {K0 | K2}, VGPR1 = {K1 | K3} split across lane halves.
    v2f bop;
    bop.x = hi ? (-2.0f * qz) : (-2.0f * qx);
    bop.y = hi ? 1.0f         : (-2.0f * qy);

    // C operand: |q|^2 for this query column, all 8 row-VGPRs.
    v8f cin;
#pragma unroll
    for (int rr = 0; rr < 8; ++rr) cin[rr] = qsq;

    int cnt = 0;
    int i0 = qg, i1 = 0, i2 = 0, i3 = 0;  // i0 defaults to self (always in ball)

    // Per-lane candidate pointer: candidate (cbase + q) and (cbase + 16 + q).
    const float* cp = vb + q * 3;

    for (int t = 0; t < NN / 32; ++t, cp += 96) {
        const int cbase = t * 32;

        // Tile 0: candidates cbase+q, tile 1: candidates cbase+16+q.
        const float cx0 = cp[0],  cy0 = cp[1],  cz0 = cp[2];
        const float cx1 = cp[48], cy1 = cp[49], cz1 = cp[50];
        __builtin_prefetch(cp + 96 * 4, 0, 0);   // global_prefetch_b8, 4 iterations (8 tiles) ahead

        const float csq0 = cx0 * cx0 + cy0 * cy0 + cz0 * cz0;
        const float csq1 = cx1 * cx1 + cy1 * cy1 + cz1 * cz1;

        v2f a0, a1;
        a0.x = hi ? cz0  : cx0;   a0.y = hi ? csq0 : cy0;
        a1.x = hi ? cz1  : cx1;   a1.y = hi ? csq1 : cy1;

        // Two independent WMMAs (same B, same C) -> pipelined.
        v8f d0 = __builtin_amdgcn_wmma_f32_16x16x4_f32(
            false, a0, false, bop, (short)0, cin, false, false);
        v8f d1 = __builtin_amdgcn_wmma_f32_16x16x4_f32(
            false, a1, false, bop, (short)0, cin, false, false);

        // 8 in-ball bits per WMMA for this half; tile0 -> [7:0], tile1 -> [23:16].
        unsigned m0 = 0u, m1 = 0u;
#pragma unroll
        for (int rr = 0; rr < 8; ++rr) {
            m0 |= (d0[rr] <= r2) ? (1u << rr) : 0u;
            m1 |= (d1[rr] <= r2) ? (1u << rr) : 0u;
        }
        const unsigned packed = m0 | (m1 << 16);
        const unsigned other  = (unsigned)__shfl_xor((int)packed, 16);

        // Branchless merge: a = bits of candidates {0-7, 16-23},
        //                   b = bits of candidates {8-15, 24-31}.
        const unsigned ma  = hi ? other  : packed;
        const unsigned mb  = hi ? packed : other;
        unsigned m32 = (ma & 0x00FF00FFu) | ((mb & 0x00FF00FFu) << 8);
        // bit k of m32 <-> candidate index cbase + k, in order.

        // First-4 lowest set bits == 4 lowest-index in-ball neighbors.
        while (m32 && cnt < 4) {
            const int bit = __builtin_ctz(m32);
            const int gi  = cbase + bit;
            i0 = (cnt == 0) ? gi : i0;
            i1 = (cnt == 1) ? gi : i1;
            i2 = (cnt == 2) ? gi : i2;
            i3 = (cnt == 3) ? gi : i3;
            ++cnt;
            m32 &= m32 - 1u;
        }

        // Wave-uniform early exit (EXEC stays all-1s around the WMMAs).
        if (__builtin_amdgcn_ballot_w32(cnt >= 4) == 0xFFFFFFFFu) break;
    }

    // Pad missing slots with group_first (== i0), as the reference does.
    i1 = (cnt > 1) ? i1 : i0;
    i2 = (cnt > 2) ? i2 : i0;
    i3 = (cnt > 3) ? i3 : i0;

    if (!hi) {
        int4 v;
        v.x = i0; v.y = i1; v.z = i2; v.w = i3;
        ((int4*)nbr)[(size_t)b * QQ + j] = v;

        // vertices_pool: we already hold the gathered query xyz.
        float* ov = out_verts + ((size_t)b * QQ + j) * 3;
        ov[0] = qx; ov[1] = qy; ov[2] = qz;
    }
}

// ---------------------------------------------------------------------------
// Kernel 2: 4-neighbor feature gather + max pool. One thread = one float4
// channel chunk of one pooled point; 64 consecutive threads stream one
// neighbor row coalesced (256 channels = 1 KB per row, L2-friendly).
// ---------------------------------------------------------------------------
__global__ __launch_bounds__(256) void maxpool_kernel(
    const float* __restrict__ feat,      // BB*NN*CC
    const int*   __restrict__ nbr,       // BB*QQ*4
    float*       __restrict__ out_feat)  // BB*QQ*CC
{
    const int CH  = CC / 4;                                // float4 chunks/row
    const int idx = blockIdx.x * blockDim.x + threadIdx.x; // BB*QQ*CH total
    const int bj  = idx / CH;
    const int c4  = idx - bj * CH;
    const int b   = bj / QQ;

    const int4 nb = ((const int4*)nbr)[bj];
    const float4* fb = (const float4*)(feat + (size_t)b * NN * CC);

    const float4 a0 = fb[(size_t)nb.x * CH + c4];
    const float4 a1 = fb[(size_t)nb.y * CH + c4];
    const float4 a2 = fb[(size_t)nb.z * CH + c4];
    const float4 a3 = fb[(size_t)nb.w * CH + c4];

    float4 o;
    o.x = fmaxf(fmaxf(a0.x, a1.x), fmaxf(a2.x, a3.x));
    o.y = fmaxf(fmaxf(a0.y, a1.y), fmaxf(a2.y, a3.y));
    o.z = fmaxf(fmaxf(a0.z, a1.z), fmaxf(a2.z, a3.z));
    o.w = fmaxf(fmaxf(a0.w, a1.w), fmaxf(a2.w, a3.w));

    ((float4*)out_feat)[idx] = o;
}

extern "C" void kernel_launch(void* const* d_in, const int* in_sizes, int n_in,
                              void* d_out, int out_size, void* d_ws, size_t ws_size,
                              hipStream_t stream) {
    (void)in_sizes; (void)n_in; (void)out_size; (void)ws_size;

    const float* vertices   = (const float*)d_in[0];
    const float* feature    = (const float*)d_in[1];
    const int*   radius     = (const int*)d_in[2];
    const int*   sample_idx = (const int*)d_in[3];

    float* out_verts = (float*)d_out;                       // BB*QQ*3
    float* out_feat  = out_verts + (size_t)BB * QQ * 3;     // BB*QQ*CC
    int*   nbr       = (int*)d_ws;                          // BB*QQ*4 ints

    // Phase 1: WMMA ball query (one wave per 16 sampled points).
    qball_wmma_kernel<<<BB * QQ / QT, 32, 0, stream>>>(
        vertices, radius, sample_idx, nbr, out_verts);

    // Phase 2: neighbor feature max-pool.
    const int total = BB * QQ * (CC / 4);
    maxpool_kernel<<<total / 256, 256, 0, stream>>>(feature, nbr, out_feat);
}